// LReLu_33629593928166
// MI455X (gfx1250) — compile-verified
//
#include <hip/hip_runtime.h>

typedef float v2f __attribute__((ext_vector_type(2)));
typedef float v8f __attribute__((ext_vector_type(8)));

#define WGSIZE     128          // 4 wave32
#define TILE       32           // output tile per workgroup
#define IN_T       42           // valid input halo span
#define IN_TP      48           // padded to multiple of 16
#define IN_STR     49
#define MID_P      80           // 74 intermediate rows/cols padded to 5x16
#define SU_STR     49
#define SY_STR     81
#define ST_STR     33

#define SLOPE 0.2f
#define GAIN  1.41421356237309515f

// D = A(16x4) * B(4x16) + C, fp32 WMMA.
// Per-lane layout (wave32): l = lane&15, h = lane>>4
//   A: comp v -> (M=l, K=v+2h)   B: comp v -> (K=v+2h, N=l)
//   C/D: comp v -> (M=v+8h, N=l)
static __device__ __forceinline__ v8f wmma_f32(v2f a, v2f b, v8f c) {
  return __builtin_amdgcn_wmma_f32_16x16x4_f32(false, a, false, b, (short)0, c,
                                               false, false);
}

__global__ __launch_bounds__(WGSIZE)
void upfirdn_lrelu_kernel(const float* __restrict__ x,
                          const float* __restrict__ upf,
                          const float* __restrict__ dnf,
                          const float* __restrict__ bias,
                          float* __restrict__ out) {
  __shared__ float sIn[IN_TP * IN_STR];   // 48x49 input halo (zero-padded)
  __shared__ float sU [MID_P * SU_STR];   // up-H result: [m(0..79)][c(0..47)]
  __shared__ float sY [MID_P * SY_STR];   // after up-W + bias + lrelu: [my][mx]
  __shared__ float sT [MID_P * ST_STR];   // after down-H: [mx][oy]

  const int tid  = threadIdx.x;
  const int lane = tid & 31;
  const int l    = lane & 15;
  const int hh   = lane >> 4;
  const int wv   = tid >> 5;               // wave id 0..3

  const int bid  = blockIdx.x;
  const int img  = bid >> 4;               // 0..2047 = b*128 + c
  const int tile = bid & 15;
  const int oy0  = (tile >> 2) * TILE;
  const int ox0  = (tile & 3)  * TILE;
  const int ch   = img & 127;
  const float* xin  = x   + (size_t)img * (128 * 128);
  float*       yout = out + (size_t)img * (128 * 128);

  // ---------------- B fragments (banded filter matrices), per lane ----------
  // Up (lhs_dilation=2, 12 taps, pad 11): out[m] = sum_j f2[j]*x[(m-j)/2].
  // Polyphase: with tile base even, kappa = (n>>1)+a, a in [0,6):
  //   weight = f2[ (n&1) ? 11-2a : 10-2a ],  f2 = 2*up_filter.
  v2f Bup[4];
#pragma unroll
  for (int t = 0; t < 4; ++t) {
    const int k0 = 4 * t + 2 * hh;
    const int a0 = k0 - (l >> 1);
    const int a1 = a0 + 1;
    const float bx =
        (a0 >= 0 && a0 < 6) ? 2.0f * upf[(l & 1) ? (11 - 2 * a0) : (10 - 2 * a0)] : 0.f;
    const float by =
        (a1 >= 0 && a1 < 6) ? 2.0f * upf[(l & 1) ? (11 - 2 * a1) : (10 - 2 * a1)] : 0.f;
    Bup[t].x = bx; Bup[t].y = by;
  }
  // Down (stride 2, 12 taps, pad 0): out[o] = sum_k dn[11-k]*y[2o+k].
  // kappa = 2n + k -> B(kappa,n) = dn[11-(kappa-2n)] inside band, else 0.
  v2f Bdn[11];
#pragma unroll
  for (int t = 0; t < 11; ++t) {
    const int k0 = 4 * t + 2 * hh;
    const int a0 = k0 - 2 * l;
    const int a1 = a0 + 1;
    const float bx = (a0 >= 0 && a0 < 12) ? dnf[11 - a0] : 0.f;
    const float by = (a1 >= 0 && a1 < 12) ? dnf[11 - a1] : 0.f;
    Bdn[t].x = bx; Bdn[t].y = by;
  }
  const float bch = bias[ch];

  // ---------------- load 42x42 halo (zero-padded to 48x48) ------------------
  // Row-strided: wave wv handles rows wv, wv+4, ...  Each lane owns fixed
  // columns (lane) and (lane+32, lanes 0..15 only); column validity is
  // loop-invariant and hoisted out of the row loop.
  {
    const int gy0 = oy0 - 5, gx0 = ox0 - 5;
    const int c0  = lane;
    const int c1  = lane + 32;
    const int gxa = gx0 + c0;
    const int gxb = gx0 + c1;
    const bool cok0 = (c0 < IN_T) && (gxa >= 0) && (gxa < 128);
    const bool cok1 = (c1 < IN_T) && (gxb >= 0) && (gxb < 128);
    for (int r = wv; r < IN_TP; r += 4) {
      const int gy = gy0 + r;
      const bool rok = (r < IN_T) && (gy >= 0) && (gy < 128);
      const float* row = xin + gy * 128;
      float v0 = 0.f, v1 = 0.f;
      if (rok && cok0) v0 = row[gxa];
      if (rok && cok1) v1 = row[gxb];
      sIn[r * IN_STR + c0] = v0;
      if (c1 < IN_TP) sIn[r * IN_STR + c1] = v1;
    }
  }
  __syncthreads();

  // ---------------- pass U1: up along H  (sIn -> sU) ------------------------
  // GEMM: M = input col, N = intermediate row m, K = input-row window (16).
  for (int s = wv; s < 15; s += 4) {           // 3 c-tiles x 5 m-tiles
    const int cBase = (s % 3) * 16;
    const int mBase = (s / 3) * 16;
    const int rBase = mBase >> 1;
    v8f acc = {};
#pragma unroll
    for (int t = 0; t < 4; ++t) {
      const int rr = rBase + 4 * t + 2 * hh;
      v2f a;
      a.x = sIn[rr * IN_STR + cBase + l];
      a.y = sIn[(rr + 1) * IN_STR + cBase + l];
      acc = wmma_f32(a, Bup[t], acc);
    }
#pragma unroll
    for (int v = 0; v < 8; ++v)
      sU[(mBase + l) * SU_STR + cBase + v + 8 * hh] = acc[v];
  }
  __syncthreads();

  // ---------------- pass U2: up along W + bias + lrelu (sU -> sY) -----------
  // GEMM: M = row m, N = intermediate col mx, K = sU-col window (16).
  for (int s = wv; s < 25; s += 4) {           // 5 m-tiles x 5 mx-tiles
    const int mBase  = (s / 5) * 16;
    const int nBase  = (s % 5) * 16;
    const int cBase  = nBase >> 1;
    v8f acc = {};
#pragma unroll
    for (int t = 0; t < 4; ++t) {
      const int cc = cBase + 4 * t + 2 * hh;
      v2f a;
      a.x = sU[(mBase + l) * SU_STR + cc];
      a.y = sU[(mBase + l) * SU_STR + cc + 1];
      acc = wmma_f32(a, Bup[t], acc);
    }
#pragma unroll
    for (int v = 0; v < 8; ++v) {
      float z = acc[v] + bch;
      z = (z >= 0.f ? z : z * SLOPE) * GAIN;
      sY[(mBase + v + 8 * hh) * SY_STR + nBase + l] = z;
    }
  }
  __syncthreads();

  // ---------------- pass D1: down along H (sY -> sT) ------------------------
  // GEMM: M = mx, N = output row oy, K = sY-row window (44, band-zeroed).
  for (int s = wv; s < 10; s += 4) {           // 5 mx-tiles x 2 oy-tiles
    const int mxBase  = (s % 5) * 16;
    const int oyBase  = (s / 5) * 16;
    const int colBase = 2 * oyBase;
    v8f acc = {};
#pragma unroll
    for (int t = 0; t < 11; ++t) {
      const int rr = colBase + 4 * t + 2 * hh;
      v2f a;
      a.x = sY[rr * SY_STR + mxBase + l];
      a.y = sY[(rr + 1) * SY_STR + mxBase + l];
      acc = wmma_f32(a, Bdn[t], acc);
    }
#pragma unroll
    for (int v = 0; v < 8; ++v)
      sT[(mxBase + v + 8 * hh) * ST_STR + oyBase + l] = acc[v];
  }
  __syncthreads();

  // ---------------- pass D2: down along W (sT -> out) -----------------------
  // One 16x16 output subtile per wave (wv in [0,4)).
  {
    const int oyBase  = (wv >> 1) * 16;
    const int oxBase  = (wv & 1) * 16;
    const int colBase = 2 * oxBase;
    v8f acc = {};
#pragma unroll
    for (int t = 0; t < 11; ++t) {
      const int rr = colBase + 4 * t + 2 * hh;
      v2f a;
      a.x = sT[rr * ST_STR + oyBase + l];
      a.y = sT[(rr + 1) * ST_STR + oyBase + l];
      acc = wmma_f32(a, Bdn[t], acc);
    }
#pragma unroll
    for (int v = 0; v < 8; ++v)
      yout[(oy0 + oyBase + v + 8 * hh) * 128 + ox0 + oxBase + l] = acc[v];
  }
}

extern "C" void kernel_launch(void* const* d_in, const int* in_sizes, int n_in,
                              void* d_out, int out_size, void* d_ws, size_t ws_size,
                              hipStream_t stream) {
  const float* x    = (const float*)d_in[0];
  const float* upf  = (const float*)d_in[1];
  const float* dnf  = (const float*)d_in[2];
  const float* bias = (const float*)d_in[3];
  float* out = (float*)d_out;
  // 2048 images x (4x4) tiles of 32x32 outputs
  upfirdn_lrelu_kernel<<<dim3(2048 * 16), dim3(WGSIZE), 0, stream>>>(
      x, upf, dnf, bias, out);
}